// PretrainableHeteroGNN_74921409511555
// MI455X (gfx1250) — compile-verified
//
#include <hip/hip_runtime.h>
#include <math.h>
#include <stdint.h>

// ---------------- problem constants (match reference) ----------------
#define NE   60000
#define NA   30000
#define HID  128
#define NH   4
#define HD   32          // per-head dim in HGT
#define NL   2
#define E_EE 150000
#define E_EA 75000
#define E_AE 75000

typedef float v2f __attribute__((ext_vector_type(2)));
typedef float v4f __attribute__((ext_vector_type(4)));
typedef float v8f __attribute__((ext_vector_type(8)));

// ---------------- small device helpers ----------------
__device__ inline float gelu_exact(float x) {
  return 0.5f * x * (1.0f + erff(x * 0.70710678118654752f));
}

__device__ inline void atomic_max_float(float* addr, float val) {
  // int-max for non-negative, uint-min for negative; init must be -inf
  if (val >= 0.0f) atomicMax((int*)addr, __float_as_int(val));
  else             atomicMin((unsigned int*)addr, __float_as_uint(val));
}

__device__ inline float wave_sum32(float v) {
#pragma unroll
  for (int off = 16; off > 0; off >>= 1) v += __shfl_xor(v, off, 32);
  return v;
}

__device__ inline void wait_async0() {
#if __has_builtin(__builtin_amdgcn_s_wait_asynccnt)
  __builtin_amdgcn_s_wait_asynccnt(0);
#else
  asm volatile("s_wait_asynccnt 0" ::: "memory");
#endif
}

// ---------------- fast WMMA fp32 GEMM (A,B row-major, K<=128, N%64==0) ----
// Workgroup = 8 waves covering 8 consecutive 16-row M tiles x one 64-col N group.
// B panel (K x 64) is staged into LDS with CDNA5 async-to-LDS loads (ASYNCcnt),
// then each wave runs 4 accumulators (16x64) off one b64 A fragment per K-step.
#define FAST_KMAX 128

__global__ void wmma_gemm_fast_kernel(const float* __restrict__ A,
                                      const float* __restrict__ B,
                                      const float* __restrict__ bias,
                                      float* __restrict__ C,
                                      int M, int K,
                                      long lda, long ldb, long ldc,
                                      long a_bs, long b_bs, long c_bs,
                                      int ngroups, int act) {
  __shared__ float Bs[FAST_KMAX * 64];

  int mg = blockIdx.x / ngroups;
  int ng = blockIdx.x - mg * ngroups;
  int n0 = ng << 6;

  const float* Ab = A + (long)blockIdx.z * a_bs;
  const float* Bb = B + (long)blockIdx.z * b_bs;
  float*       Cb = C + (long)blockIdx.z * c_bs;

  // ---- async cooperative load of B panel rows 0..K-1, cols n0..n0+63 ----
  {
    int total = K * 64;                        // multiple of 1024
    for (int idx = (int)threadIdx.x * 4; idx < total; idx += 256 * 4) {
      int row = idx >> 6, col = idx & 63;
      unsigned lds_addr = (unsigned)(size_t)(&Bs[idx]);
      unsigned long long gaddr =
          (unsigned long long)(size_t)(Bb + (long)row * ldb + n0 + col);
      asm volatile("global_load_async_to_lds_b128 %0, %1, off"
                   :: "v"(lds_addr), "v"(gaddr) : "memory");
    }
    wait_async0();
  }
  __syncthreads();

  int wave = threadIdx.x >> 5;
  int lane = threadIdx.x & 31;
  int half = lane >> 4;
  int lm   = lane & 15;
  int mt   = mg * 8 + wave;
  if (mt * 16 >= M) return;                    // after barrier: safe

  const float* Ap = Ab + (long)(mt * 16 + lm) * lda;
  v8f acc0 = {}, acc1 = {}, acc2 = {}, acc3 = {};

  for (int k = 0; k < K; k += 4) {
    int ka = k + half * 2;
    v2f a = *(const v2f*)(Ap + ka);            // contiguous b64 load
    const float* bs0 = &Bs[ka * 64 + lm];
    const float* bs1 = &Bs[(ka + 1) * 64 + lm];
    v2f b;
    b.x = bs0[0];  b.y = bs1[0];
    acc0 = __builtin_amdgcn_wmma_f32_16x16x4_f32(false, a, false, b, (short)0, acc0, false, false);
    b.x = bs0[16]; b.y = bs1[16];
    acc1 = __builtin_amdgcn_wmma_f32_16x16x4_f32(false, a, false, b, (short)0, acc1, false, false);
    b.x = bs0[32]; b.y = bs1[32];
    acc2 = __builtin_amdgcn_wmma_f32_16x16x4_f32(false, a, false, b, (short)0, acc2, false, false);
    b.x = bs0[48]; b.y = bs1[48];
    acc3 = __builtin_amdgcn_wmma_f32_16x16x4_f32(false, a, false, b, (short)0, acc3, false, false);
  }

  v8f accs[4] = {acc0, acc1, acc2, acc3};
#pragma unroll
  for (int j = 0; j < 4; j++) {
    long ncol = n0 + j * 16 + lm;
    float bv = bias ? bias[ncol] : 0.0f;
#pragma unroll
    for (int i = 0; i < 8; i++) {
      long row = (long)(mt * 16 + half * 8 + i);   // D layout: vgpr i -> M=i / M=8+i
      float v = accs[j][i] + bv;
      if (act == 1) v = fmaxf(v, 0.0f);
      Cb[row * ldc + ncol] = v;
    }
  }
}

// ---------------- generic strided/batched/split-K WMMA fp32 GEMM ----------
// C[b][m, n] = act( sum_k A[b][m,k] * B[b][k,n] + bias[n] )
// A element at A + b*a_bs + m*a_rs + k*a_cs   (supports transposed access)
// B element at B + b*b_bs + k*b_rs + n*b_cs
__global__ void wmma_gemm_kernel(const float* __restrict__ A,
                                 const float* __restrict__ B,
                                 const float* __restrict__ bias,
                                 float* __restrict__ C,
                                 int M, int N,
                                 long a_rs, long a_cs, long b_rs, long b_cs,
                                 long ldc, long a_bs, long b_bs, long c_bs,
                                 int act, int kper, int atomicC) {
  int tiles_n = N >> 4;
  int tiles   = (M >> 4) * tiles_n;
  int tid = blockIdx.x * (blockDim.x >> 5) + (threadIdx.x >> 5);
  if (tid >= tiles) return;                       // wave-uniform, EXEC stays full
  int mt = tid / tiles_n;
  int nt = tid - mt * tiles_n;

  const float* Ab = A + (long)blockIdx.z * a_bs;
  const float* Bb = B + (long)blockIdx.z * b_bs;
  float*       Cb = C + (long)blockIdx.z * c_bs;

  int lane = threadIdx.x & 31;
  int half = lane >> 4;
  int lm   = lane & 15;
  long arow = (long)(mt * 16 + lm);               // A: lanes encode M
  long ncol = (long)(nt * 16 + lm);               // B/C: lanes encode N

  int k0 = blockIdx.y * kper;
  int k1 = k0 + kper;

  v8f acc = {};
#pragma unroll 2
  for (int k = k0; k < k1; k += 4) {
    int ka = k + half * 2;
    v2f a, b;
    a.x = Ab[arow * a_rs + (long)ka       * a_cs];
    a.y = Ab[arow * a_rs + (long)(ka + 1) * a_cs];
    b.x = Bb[(long)ka       * b_rs + ncol * b_cs];
    b.y = Bb[(long)(ka + 1) * b_rs + ncol * b_cs];
    acc = __builtin_amdgcn_wmma_f32_16x16x4_f32(false, a, false, b,
                                                (short)0, acc, false, false);
  }

  float bv = bias ? bias[ncol] : 0.0f;
#pragma unroll
  for (int i = 0; i < 8; i++) {
    long row = (long)(mt * 16 + half * 8 + i);
    float v = acc[i] + bv;
    if (act == 1) v = fmaxf(v, 0.0f);
    if (atomicC) atomicAdd(&Cb[row * ldc + ncol], v);
    else         Cb[row * ldc + ncol] = v;
  }
}

// ---------------- elementwise / reduction kernels ----------------
__global__ void fill_kernel(float* p, long n, float v) {
  long i = (long)blockIdx.x * blockDim.x + threadIdx.x;
  if (i < n) p[i] = v;
}

__global__ void gelu_kernel(float* x, long n) {
  long i = (long)blockIdx.x * blockDim.x + threadIdx.x;
  if (i < n) x[i] = gelu_exact(x[i]);
}

// h = gelu( g*o + (1-g)*h ),  g = sigmoid(skip[l])
__global__ void blend_gelu_kernel(float* h, const float* __restrict__ o,
                                  const float* __restrict__ skip, int l, long n) {
  long i = (long)blockIdx.x * blockDim.x + threadIdx.x;
  if (i >= n) return;
  float g = 1.0f / (1.0f + expf(-skip[l]));
  float v = g * o[i] + (1.0f - g) * h[i];
  h[i] = gelu_exact(v);
}

// out = (opt relu)( LN(ax*X + ay*Y) * g + b ), one wave per 128-wide row
__global__ void ln_kernel(const float* __restrict__ X, const float* __restrict__ Y,
                          float ax, float ay,
                          const float* __restrict__ g, const float* __restrict__ b,
                          float* __restrict__ out, int rows, int relu) {
  int wid  = blockIdx.x * (blockDim.x >> 5) + (threadIdx.x >> 5);
  int lane = threadIdx.x & 31;
  if (wid >= rows) return;
  const float* xp = X + (long)wid * HID + lane * 4;
  const float* yp = Y ? Y + (long)wid * HID + lane * 4 : nullptr;
  float v[4];
  float s = 0.0f;
#pragma unroll
  for (int j = 0; j < 4; j++) {
    float t = ax * xp[j];
    if (yp) t += ay * yp[j];
    v[j] = t; s += t;
  }
  s = wave_sum32(s);
  float mu = s * (1.0f / HID);
  float var = 0.0f;
#pragma unroll
  for (int j = 0; j < 4; j++) { float d = v[j] - mu; var += d * d; }
  var = wave_sum32(var) * (1.0f / HID);
  float r = rsqrtf(var + 1e-5f);
#pragma unroll
  for (int j = 0; j < 4; j++) {
    float o = (v[j] - mu) * r * g[lane * 4 + j] + b[lane * 4 + j];
    if (relu) o = fmaxf(o, 0.0f);
    out[(long)wid * HID + lane * 4 + j] = o;
  }
}

__global__ void sumsq_kernel(const float* __restrict__ x, long n, float* out) {
  __shared__ float sm[256];
  float acc = 0.0f;
  for (long i = (long)blockIdx.x * blockDim.x + threadIdx.x; i < n;
       i += (long)gridDim.x * blockDim.x) {
    float v = x[i]; acc += v * v;
  }
  sm[threadIdx.x] = acc; __syncthreads();
  for (int s = 128; s > 0; s >>= 1) {
    if ((int)threadIdx.x < s) sm[threadIdx.x] += sm[threadIdx.x + s];
    __syncthreads();
  }
  if (threadIdx.x == 0) atomicAdd(out, sm[0]);
}

__global__ void scale_invnorm_kernel(float* x, long n, const float* __restrict__ ss) {
  long i = (long)blockIdx.x * blockDim.x + threadIdx.x;
  if (i < n) x[i] *= rsqrtf(fmaxf(ss[0], 1e-30f));
}

__global__ void colsum_kernel(const float* __restrict__ x, int rows, int cols,
                              int rowchunk, float* out) {
  int c = blockIdx.x * blockDim.x + threadIdx.x;
  if (c >= cols) return;
  int r0 = blockIdx.y * rowchunk;
  int r1 = min(r0 + rowchunk, rows);
  float acc = 0.0f;
  for (int r = r0; r < r1; r++) acc += x[(long)r * cols + c];
  atomicAdd(&out[c], acc);
}

// ---------------- edge (message-passing) kernels ----------------
// al[e,h] = (q[dst,h,:] . kt[src,h,:]) * p_rel[h] / sqrt(D); atomic max into m[dst,h]
__global__ void edge_logits_kernel(const int* __restrict__ src, const int* __restrict__ dst,
                                   const float* __restrict__ kqv_dst,   // ld 384, q at +128
                                   const float* __restrict__ kt,        // [Nsrc, 128]
                                   const float* __restrict__ prel,
                                   float* __restrict__ al, float* __restrict__ m, int E) {
  int i = blockIdx.x * blockDim.x + threadIdx.x;
  if (i >= E * NH) return;
  int e = i >> 2, h = i & 3;
  int sN = src[e], dN = dst[e];
  const float* qp = kqv_dst + (long)dN * 384 + 128 + h * HD;
  const float* kp = kt + (long)sN * HID + h * HD;
  float acc = 0.0f;
#pragma unroll
  for (int j = 0; j < HD; j++) acc += qp[j] * kp[j];
  acc *= prel[h] * 0.17677669529663689f;   // 1/sqrt(32)
  al[i] = acc;
  atomic_max_float(&m[(long)dN * NH + h], acc);
}

__global__ void edge_exp_kernel(const int* __restrict__ dst, float* __restrict__ al,
                                const float* __restrict__ m, float* __restrict__ s, int E) {
  int i = blockIdx.x * blockDim.x + threadIdx.x;
  if (i >= E * NH) return;
  int e = i >> 2, h = i & 3;
  int dN = dst[e];
  float w = expf(al[i] - m[(long)dN * NH + h]);
  al[i] = w;
  atomicAdd(&s[(long)dN * NH + h], w);
}

__global__ void edge_agg_kernel(const int* __restrict__ src, const int* __restrict__ dst,
                                const float* __restrict__ al, const float* __restrict__ s,
                                const float* __restrict__ vt, float* __restrict__ agg, int E) {
  int i = blockIdx.x * blockDim.x + threadIdx.x;
  if (i >= E * NH) return;
  int e = i >> 2, h = i & 3;
  int sN = src[e], dN = dst[e];
  float w = al[i] / (s[(long)dN * NH + h] + 1e-16f);
  const float* vp = vt + (long)sN * HID + h * HD;
  float* ap = agg + (long)dN * HID + h * HD;
#pragma unroll
  for (int j = 0; j < HD; j++) atomicAdd(&ap[j], w * vp[j]);
}

// ---------------- SGFormer helpers ----------------
__global__ void den_kernel(const float* __restrict__ q, const float* __restrict__ knsum,
                           float* __restrict__ den, int n, float nf) {
  int i = blockIdx.x * blockDim.x + threadIdx.x;
  if (i >= n * NH) return;
  int nn = i >> 2, h = i & 3;
  const float* qp = q + (long)nn * 512 + h * HID;
  const float* kp = knsum + h * HID;
  float acc = 0.0f;
#pragma unroll 4
  for (int j = 0; j < HID; j++) acc += qp[j] * kp[j];
  den[i] = acc + nf;
}

__global__ void attn_mean_kernel(const float* __restrict__ num, const float* __restrict__ v,
                                 const float* __restrict__ den, float* __restrict__ attn,
                                 int n, float nf) {
  long i = (long)blockIdx.x * blockDim.x + threadIdx.x;
  if (i >= (long)n * HID) return;
  long nn = i >> 7;
  int d = (int)(i & 127);
  float acc = 0.0f;
#pragma unroll
  for (int h = 0; h < NH; h++) {
    float x = num[nn * 512 + h * HID + d] + nf * v[nn * 512 + h * HID + d];
    acc += x / den[nn * NH + h];
  }
  attn[i] = acc * 0.25f;
}

// ---------------- host-side GEMM launcher (dispatch fast / generic) -------
static inline void gemm(hipStream_t st, const float* A, const float* B, const float* bias,
                        float* C, int M, int N, int K,
                        long a_rs, long a_cs, long b_rs, long b_cs, long ldc,
                        long a_bs = 0, long b_bs = 0, long c_bs = 0, int batch = 1,
                        int act = 0, int ksplit = 1, int atomicC = 0) {
  if (ksplit == 1 && !atomicC && a_cs == 1 && b_cs == 1 &&
      K <= FAST_KMAX && (K & 3) == 0 && (N & 63) == 0) {
    int ngroups = N / 64;
    int mgroups = (M / 16 + 7) / 8;
    dim3 grid((unsigned)(mgroups * ngroups), 1, (unsigned)batch);
    wmma_gemm_fast_kernel<<<grid, dim3(256), 0, st>>>(A, B, bias, C, M, K,
                                                      a_rs, b_rs, ldc,
                                                      a_bs, b_bs, c_bs, ngroups, act);
  } else {
    int tiles = (M / 16) * (N / 16);
    dim3 grid((unsigned)((tiles + 7) / 8), (unsigned)ksplit, (unsigned)batch);
    wmma_gemm_kernel<<<grid, dim3(256), 0, st>>>(A, B, bias, C, M, N,
                                                 a_rs, a_cs, b_rs, b_cs, ldc,
                                                 a_bs, b_bs, c_bs, act, K / ksplit, atomicC);
  }
}

static inline int cdiv(long a, long b) { return (int)((a + b - 1) / b); }

// ---------------- entry point ----------------
extern "C" void kernel_launch(void* const* d_in, const int* in_sizes, int n_in,
                              void* d_out, int out_size, void* d_ws, size_t ws_size,
                              hipStream_t stream) {
  (void)in_sizes; (void)n_in; (void)out_size; (void)ws_size;

  const float* x_entity = (const float*)d_in[0];
  const float* x_attr   = (const float*)d_in[1];
  const float* Wkqv_ent = (const float*)d_in[2];
  const float* bkqv_ent = (const float*)d_in[3];
  const float* Wkqv_att = (const float*)d_in[4];
  const float* bkqv_att = (const float*)d_in[5];
  const float* Wout_ent = (const float*)d_in[6];
  const float* bout_ent = (const float*)d_in[7];
  const float* Wout_att = (const float*)d_in[8];
  const float* bout_att = (const float*)d_in[9];
  const float* skip_ent = (const float*)d_in[10];
  const float* skip_att = (const float*)d_in[11];
  const float* a_rel    = (const float*)d_in[12];
  const float* m_rel    = (const float*)d_in[13];
  const float* p_rel    = (const float*)d_in[14];
  const float* ln_ent_g = (const float*)d_in[15];
  const float* ln_ent_b = (const float*)d_in[16];
  const float* sg_fc_W  = (const float*)d_in[17];
  const float* sg_fc_b  = (const float*)d_in[18];
  const float* sg_ln0_g = (const float*)d_in[19];
  const float* sg_ln0_b = (const float*)d_in[20];
  const float* sg_Wq    = (const float*)d_in[21];
  const float* sg_bq    = (const float*)d_in[22];
  const float* sg_Wk    = (const float*)d_in[23];
  const float* sg_bk    = (const float*)d_in[24];
  const float* sg_Wv    = (const float*)d_in[25];
  const float* sg_bv    = (const float*)d_in[26];
  const float* sg_ln1_g = (const float*)d_in[27];
  const float* sg_ln1_b = (const float*)d_in[28];
  const float* proj_W1  = (const float*)d_in[29];
  const float* proj_b1  = (const float*)d_in[30];
  const float* proj_W2  = (const float*)d_in[31];
  const float* proj_b2  = (const float*)d_in[32];
  const int*   ei_ee    = (const int*)d_in[33];
  const int*   ei_ea    = (const int*)d_in[34];
  const int*   ei_ae    = (const int*)d_in[35];

  // ------- workspace layout (floats), two overlapping arenas -------
  float* W = (float*)d_ws;
  size_t o = 0;
  auto alloc = [&](size_t n) { float* p = W + o; o += n; return p; };

  float* HE = alloc((size_t)NE * HID);   // persistent node states
  float* HA = alloc((size_t)NA * HID);
  size_t arena = o;

  // HGT-layer arena
  float* KQVE = alloc((size_t)NE * 384);
  float* KQVA = alloc((size_t)NA * 384);
  float* KTEE = alloc((size_t)NE * HID);
  float* VTEE = alloc((size_t)NE * HID);
  float* KTEA = alloc((size_t)NE * HID);
  float* VTEA = alloc((size_t)NE * HID);
  float* KTAE = alloc((size_t)NA * HID);
  float* VTAE = alloc((size_t)NA * HID);
  float* ALEE = alloc((size_t)E_EE * NH);
  float* ALEA = alloc((size_t)E_EA * NH);
  float* ALAE = alloc((size_t)E_AE * NH);
  float* ME   = alloc((size_t)NE * NH);
  float* SE   = alloc((size_t)NE * NH);
  float* MA   = alloc((size_t)NA * NH);
  float* SA   = alloc((size_t)NA * NH);
  float* AGGE = alloc((size_t)NE * HID);
  float* AGGA = alloc((size_t)NA * HID);
  float* TMPE = alloc((size_t)NE * HID);
  float* TMPA = alloc((size_t)NA * HID);

  // SGFormer arena (reuses the layer arena)
  o = arena;
  float* Z0    = alloc((size_t)NE * HID);
  float* TMPZ  = alloc((size_t)NE * HID);
  float* Qb    = alloc((size_t)NE * 512);
  float* Kb    = alloc((size_t)NE * 512);
  float* Vb    = alloc((size_t)NE * 512);
  float* KVS   = alloc((size_t)NH * HID * HID);
  float* KNSUM = alloc((size_t)NH * HID);
  float* DEN   = alloc((size_t)NE * NH);
  float* ATTN  = alloc((size_t)NE * HID);
  float* HG    = alloc((size_t)NE * HID);
  float* HENT  = alloc((size_t)NE * HID);
  float* PT    = alloc((size_t)NE * HID);
  float* SC    = alloc(4);
  float* NUM   = Kb;   // K buffer is dead after kvs/knsum -> reuse for num

  // ------- init node states -------
  hipMemcpyAsync(HE, x_entity, sizeof(float) * (size_t)NE * HID,
                 hipMemcpyDeviceToDevice, stream);
  hipMemcpyAsync(HA, x_attr, sizeof(float) * (size_t)NA * HID,
                 hipMemcpyDeviceToDevice, stream);

  // ================= HGT layers =================
  for (int l = 0; l < NL; l++) {
    // kqv = x @ Wkqv + b   (layout: [N, 3*H*D], k|q|v at col 0|128|256)
    gemm(stream, HE, Wkqv_ent + (long)l * HID * 384, bkqv_ent + (long)l * 384, KQVE,
         NE, 384, HID, HID, 1, 384, 1, 384);
    gemm(stream, HA, Wkqv_att + (long)l * HID * 384, bkqv_att + (long)l * 384, KQVA,
         NA, 384, HID, HID, 1, 384, 1, 384);

    const float* ar[3], *mr[3];
    for (int t = 0; t < 3; t++) {
      ar[t] = a_rel + (long)(l * 3 + t) * NH * HD * HD;
      mr[t] = m_rel + (long)(l * 3 + t) * NH * HD * HD;
    }
    // per-head D x D relation transforms (batched GEMM over heads)
    gemm(stream, KQVE + 0,   ar[0], nullptr, KTEE, NE, HD, HD, 384, 1, HD, 1, HID, HD, HD * HD, HD, NH);
    gemm(stream, KQVE + 256, mr[0], nullptr, VTEE, NE, HD, HD, 384, 1, HD, 1, HID, HD, HD * HD, HD, NH);
    gemm(stream, KQVE + 0,   ar[1], nullptr, KTEA, NE, HD, HD, 384, 1, HD, 1, HID, HD, HD * HD, HD, NH);
    gemm(stream, KQVE + 256, mr[1], nullptr, VTEA, NE, HD, HD, 384, 1, HD, 1, HID, HD, HD * HD, HD, NH);
    gemm(stream, KQVA + 0,   ar[2], nullptr, KTAE, NA, HD, HD, 384, 1, HD, 1, HID, HD, HD * HD, HD, NH);
    gemm(stream, KQVA + 256, mr[2], nullptr, VTAE, NA, HD, HD, 384, 1, HD, 1, HID, HD, HD * HD, HD, NH);

    // softmax state + aggregation buffers
    fill_kernel<<<cdiv((long)NE * NH, 256), 256, 0, stream>>>(ME, (long)NE * NH, -INFINITY);
    fill_kernel<<<cdiv((long)NE * NH, 256), 256, 0, stream>>>(SE, (long)NE * NH, 0.0f);
    fill_kernel<<<cdiv((long)NA * NH, 256), 256, 0, stream>>>(MA, (long)NA * NH, -INFINITY);
    fill_kernel<<<cdiv((long)NA * NH, 256), 256, 0, stream>>>(SA, (long)NA * NH, 0.0f);
    fill_kernel<<<cdiv((long)NE * HID, 256), 256, 0, stream>>>(AGGE, (long)NE * HID, 0.0f);
    fill_kernel<<<cdiv((long)NA * HID, 256), 256, 0, stream>>>(AGGA, (long)NA * HID, 0.0f);

    const float* pr = p_rel + (long)l * 3 * NH;
    // pass 1: logits + segment max (joint over all relations per dst type)
    edge_logits_kernel<<<cdiv((long)E_EE * NH, 256), 256, 0, stream>>>(
        ei_ee, ei_ee + E_EE, KQVE, KTEE, pr + 0 * NH, ALEE, ME, E_EE);
    edge_logits_kernel<<<cdiv((long)E_EA * NH, 256), 256, 0, stream>>>(
        ei_ea, ei_ea + E_EA, KQVA, KTEA, pr + 1 * NH, ALEA, MA, E_EA);
    edge_logits_kernel<<<cdiv((long)E_AE * NH, 256), 256, 0, stream>>>(
        ei_ae, ei_ae + E_AE, KQVE, KTAE, pr + 2 * NH, ALAE, ME, E_AE);
    // pass 2: exp + segment sum
    edge_exp_kernel<<<cdiv((long)E_EE * NH, 256), 256, 0, stream>>>(ei_ee + E_EE, ALEE, ME, SE, E_EE);
    edge_exp_kernel<<<cdiv((long)E_EA * NH, 256), 256, 0, stream>>>(ei_ea + E_EA, ALEA, MA, SA, E_EA);
    edge_exp_kernel<<<cdiv((long)E_AE * NH, 256), 256, 0, stream>>>(ei_ae + E_AE, ALAE, ME, SE, E_AE);
    // pass 3: weighted scatter-add of transformed values
    edge_agg_kernel<<<cdiv((long)E_EE * NH, 256), 256, 0, stream>>>(ei_ee, ei_ee + E_EE, ALEE, SE, VTEE, AGGE, E_EE);
    edge_agg_kernel<<<cdiv((long)E_EA * NH, 256), 256, 0, stream>>>(ei_ea, ei_ea + E_EA, ALEA, SA, VTEA, AGGA, E_EA);
    edge_agg_kernel<<<cdiv((long)E_AE * NH, 256), 256, 0, stream>>>(ei_ae, ei_ae + E_AE, ALAE, SE, VTAE, AGGE, E_AE);

    // out projection: gelu(agg) @ Wout + bout, then sigmoid-skip blend + outer gelu
    gelu_kernel<<<cdiv((long)NE * HID, 256), 256, 0, stream>>>(AGGE, (long)NE * HID);
    gelu_kernel<<<cdiv((long)NA * HID, 256), 256, 0, stream>>>(AGGA, (long)NA * HID);
    gemm(stream, AGGE, Wout_ent + (long)l * HID * HID, bout_ent + (long)l * HID, TMPE,
         NE, HID, HID, HID, 1, HID, 1, HID);
    gemm(stream, AGGA, Wout_att + (long)l * HID * HID, bout_att + (long)l * HID, TMPA,
         NA, HID, HID, HID, 1, HID, 1, HID);
    blend_gelu_kernel<<<cdiv((long)NE * HID, 256), 256, 0, stream>>>(HE, TMPE, skip_ent, l, (long)NE * HID);
    blend_gelu_kernel<<<cdiv((long)NA * HID, 256), 256, 0, stream>>>(HA, TMPA, skip_att, l, (long)NA * HID);
  }

  // ================= SGFormer global attention on entities =================
  gemm(stream, HE, sg_fc_W, sg_fc_b, TMPZ, NE, HID, HID, HID, 1, HID, 1, HID);
  ln_kernel<<<cdiv(NE, 8), 256, 0, stream>>>(TMPZ, nullptr, 1.0f, 0.0f,
                                             sg_ln0_g, sg_ln0_b, Z0, NE, 1);
  gemm(stream, Z0, sg_Wq, sg_bq, Qb, NE, 512, HID, HID, 1, 512, 1, 512);
  gemm(stream, Z0, sg_Wk, sg_bk, Kb, NE, 512, HID, HID, 1, 512, 1, 512);
  gemm(stream, Z0, sg_Wv, sg_bv, Vb, NE, 512, HID, HID, 1, 512, 1, 512);

  // global Frobenius norms of q and k; scale in place (qs, kn)
  fill_kernel<<<1, 32, 0, stream>>>(SC, 4, 0.0f);
  sumsq_kernel<<<1024, 256, 0, stream>>>(Qb, (long)NE * 512, SC + 0);
  sumsq_kernel<<<1024, 256, 0, stream>>>(Kb, (long)NE * 512, SC + 1);
  scale_invnorm_kernel<<<cdiv((long)NE * 512, 256), 256, 0, stream>>>(Qb, (long)NE * 512, SC + 0);
  scale_invnorm_kernel<<<cdiv((long)NE * 512, 256), 256, 0, stream>>>(Kb, (long)NE * 512, SC + 1);

  // kn.sum(0)  -> KNSUM[h*128 + m]
  fill_kernel<<<2, 256, 0, stream>>>(KNSUM, NH * HID, 0.0f);
  {
    dim3 cg(2, 60);
    colsum_kernel<<<cg, 256, 0, stream>>>(Kb, NE, 512, 1000, KNSUM);
  }

  // kvs[h] = kn[:,h,:]^T @ v[:,h,:]  (transposed-A strided GEMM, split-K=50, atomic acc)
  fill_kernel<<<cdiv((long)NH * HID * HID, 256), 256, 0, stream>>>(KVS, (long)NH * HID * HID, 0.0f);
  gemm(stream, Kb, Vb, nullptr, KVS, HID, HID, NE,
       /*a_rs*/ 1, /*a_cs*/ 512, /*b_rs*/ 512, /*b_cs*/ 1, /*ldc*/ HID,
       /*a_bs*/ HID, /*b_bs*/ HID, /*c_bs*/ (long)HID * HID, NH,
       /*act*/ 0, /*ksplit*/ 50, /*atomicC*/ 1);

  // num[n,h,:] = qs[n,h,:] @ kvs[h]   (batched over heads; +n*v folded into attn kernel)
  gemm(stream, Qb, KVS, nullptr, NUM, NE, HID, HID,
       512, 1, HID, 1, 512, HID, (long)HID * HID, HID, NH);

  den_kernel<<<cdiv((long)NE * NH, 256), 256, 0, stream>>>(Qb, KNSUM, DEN, NE, (float)NE);
  attn_mean_kernel<<<cdiv((long)NE * HID, 256), 256, 0, stream>>>(NUM, Vb, DEN, ATTN, NE, (float)NE);

  // h_global = LN(0.5*attn + 0.5*z0);  h_ent = LN(0.9*h_e + 0.1*h_global)
  ln_kernel<<<cdiv(NE, 8), 256, 0, stream>>>(ATTN, Z0, 0.5f, 0.5f, sg_ln1_g, sg_ln1_b, HG, NE, 0);
  ln_kernel<<<cdiv(NE, 8), 256, 0, stream>>>(HE, HG, 0.9f, 0.1f, ln_ent_g, ln_ent_b, HENT, NE, 0);

  // projector MLP -> d_out
  gemm(stream, HENT, proj_W1, proj_b1, PT, NE, HID, HID, HID, 1, HID, 1, HID,
       0, 0, 0, 1, /*act=relu*/ 1);
  gemm(stream, PT, proj_W2, proj_b2, (float*)d_out, NE, HID, HID, HID, 1, HID, 1, HID);
}